// AngularPenaltySM_670014898756
// MI455X (gfx1250) — compile-verified
//
#include <hip/hip_runtime.h>
#include <hip/hip_bf16.h>
#include <math.h>

typedef __attribute__((ext_vector_type(2))) float v2f;
typedef __attribute__((ext_vector_type(8))) float v8f;

#define N_ROWS   2048
#define D_DIM    512
#define C_DIM    50000
#define S_SCALE  64.0f
#define EPS_CLIP 1e-7f
#define COS_M    0.87758256189f   /* cos(0.5) */
#define SIN_M    0.47942553860f   /* sin(0.5) */

// GEMM tiling: block = 256 threads = 8 waves (wave32), arranged 2 (rows) x 4 (cols)
// of 16x16 WMMA tiles -> 32 rows x 64 cols per block. K staged in LDS chunks of 32.
#define BK 32
#define BN 32
#define BC 64
#define AS_STRIDE 36   /* 144 B row stride: 16-B aligned, bank-spread */
#define BS_STRIDE 36

// CDNA5 async Global->LDS copy (ASYNCcnt path): LDS byte offset in a VGPR,
// 64-bit global address in a VGPR pair, copy engine writes LDS directly.
#define ASYNC_LOAD_B128(ldsoff, gaddr)                                   \
    asm volatile("global_load_async_to_lds_b128 %0, %1, off"             \
                 :: "v"(ldsoff), "v"(gaddr) : "memory")
#define WAIT_ASYNCCNT0() asm volatile("s_wait_asynccnt 0x0" ::: "memory")

__global__ void init_rowsum_kernel(float* __restrict__ rowsum) {
    int i = blockIdx.x * blockDim.x + threadIdx.x;
    if (i < N_ROWS) rowsum[i] = 0.0f;
}

// One wave per row: inv_norm[n] = 1/||x[n]||
__global__ void rownorm_kernel(const float* __restrict__ x,
                               float* __restrict__ inv_norm) {
    int row  = (blockIdx.x * blockDim.x + threadIdx.x) >> 5;
    int lane = threadIdx.x & 31;
    if (row >= N_ROWS) return;
    const float* xr = x + (size_t)row * D_DIM;
    float s = 0.0f;
    for (int k = lane; k < D_DIM; k += 32) {
        float v = xr[k];
        s = fmaf(v, v, s);
    }
    #pragma unroll
    for (int off = 16; off > 0; off >>= 1) s += __shfl_xor(s, off, 32);
    if (lane == 0) inv_norm[row] = rsqrtf(s);
}

// One wave per row: tgt[n] = inv_norm[n] * dot(x[n], W[label[n]])
__global__ void target_kernel(const float* __restrict__ x,
                              const float* __restrict__ W,
                              const int* __restrict__ labels,
                              const float* __restrict__ inv_norm,
                              float* __restrict__ tgt) {
    int row  = (blockIdx.x * blockDim.x + threadIdx.x) >> 5;
    int lane = threadIdx.x & 31;
    if (row >= N_ROWS) return;
    const float* xr = x + (size_t)row * D_DIM;
    const float* wr = W + (size_t)labels[row] * D_DIM;
    float s = 0.0f;
    for (int k = lane; k < D_DIM; k += 32) s = fmaf(xr[k], wr[k], s);
    #pragma unroll
    for (int off = 16; off > 0; off >>= 1) s += __shfl_xor(s, off, 32);
    if (lane == 0) tgt[row] = s * inv_norm[row];
}

// Fused GEMM (xn @ W^T) + exp row-sum. Never materializes the [N,C] logits.
// Staging via CDNA5 async Global->LDS copies; row-normalization folded into
// the A-fragment (VALU mults co-execute with the WMMA pipe).
__global__ __launch_bounds__(256) void gemm_exp_kernel(
        const float* __restrict__ x, const float* __restrict__ W,
        const float* __restrict__ inv_norm, float* __restrict__ rowsum) {
    __shared__ float As[BN * AS_STRIDE];
    __shared__ float Bs[BC * BS_STRIDE];
    __shared__ float rs[BN];

    const int tid  = threadIdx.x;
    const int lane = tid & 31;
    const int wave = tid >> 5;
    const int nt   = wave >> 2;          // 0..1  row-tile of this wave
    const int ct   = wave & 3;           // 0..3  col-tile of this wave
    const int nbase = blockIdx.y * BN;
    const int cbase = blockIdx.x * BC;

    if (tid < BN) rs[tid] = 0.0f;

    // Cooperative A copy mapping: 32 rows x 32 k = 4 KB -> one b128 per thread
    const int a_row = tid >> 3;          // 0..31
    const int a_k   = (tid & 7) * 4;     // 0,4,..,28
    const unsigned long long a_src =
        (unsigned long long)(x + (size_t)(nbase + a_row) * D_DIM + a_k);
    const unsigned a_lds = (unsigned)(size_t)&As[a_row * AS_STRIDE + a_k];

    // Cooperative B copy mapping: 64 cols x 32 k = 8 KB -> two b128 per thread.
    // Out-of-range columns are CLAMPED (no divergence); epilogue masks them.
    const int b_col = tid >> 2;          // 0..63
    const int b_k   = (tid & 3) * 8;     // 0,8,16,24
    const int b_colg = min(cbase + b_col, C_DIM - 1);
    const unsigned long long b_src =
        (unsigned long long)(W + (size_t)b_colg * D_DIM + b_k);
    const unsigned b_lds = (unsigned)(size_t)&Bs[b_col * BS_STRIDE + b_k];

    // Per-lane row scale for the A fragment (lane&15 = row within wave tile)
    const float ascl = inv_norm[nbase + nt * 16 + (lane & 15)];

    v8f acc = {};

    for (int k0 = 0; k0 < D_DIM; k0 += BK) {
        __syncthreads();                      // prior tile fully consumed
        const unsigned long long koff = (unsigned long long)k0 * 4ull;
        ASYNC_LOAD_B128(a_lds, a_src + koff);
        ASYNC_LOAD_B128(b_lds, b_src + koff);
        ASYNC_LOAD_B128(b_lds + 16u, b_src + koff + 16ull);
        // Prefetch next K-slab of W into L2 while the copy engine works
        if ((k0 + BK) < D_DIM)
            __builtin_prefetch((const float*)(b_src + koff) + BK, 0, 1);
        WAIT_ASYNCCNT0();                     // this wave's copies landed
        __syncthreads();                      // all waves' copies landed

        // WMMA fragment addressing (ISA 7.12.2, 32-bit A 16x4 / B 4x16):
        //   lane&15 selects row (A) / col (B); lanes 16-31 hold K+2,K+3.
        const int krel = (lane >> 4) * 2;
        const float* arow = &As[(nt * 16 + (lane & 15)) * AS_STRIDE + krel];
        const float* brow = &Bs[(ct * 16 + (lane & 15)) * BS_STRIDE + krel];
        #pragma unroll
        for (int kk = 0; kk < BK; kk += 4) {
            v2f a = *(const v2f*)(arow + kk) * ascl;   // fold L2-normalization in
            v2f b = *(const v2f*)(brow + kk);
            acc = __builtin_amdgcn_wmma_f32_16x16x4_f32(
                false, a, false, b, (short)0, acc, false, false);
        }
    }

    // Epilogue: e = exp(S*wf) masked to valid columns, reduce 16 cols per row,
    // accumulate into LDS then one global float atomic per row per block.
    const int half = lane >> 4;          // 0/1 -> rows i / i+8
    const int l16  = lane & 15;          // column within wave tile
    const bool cok = (cbase + ct * 16 + l16) < C_DIM;
    #pragma unroll
    for (int i = 0; i < 8; ++i) {
        float e = cok ? __expf(S_SCALE * acc[i]) : 0.0f;
        #pragma unroll
        for (int off = 1; off < 16; off <<= 1) e += __shfl_xor(e, off, 32);
        if (l16 == 0) atomicAdd(&rs[nt * 16 + half * 8 + i], e);
    }
    __syncthreads();
    if (tid < BN) atomicAdd(&rowsum[nbase + tid], rs[tid]);
}

__global__ void finalize_kernel(const float* __restrict__ rowsum,
                                const float* __restrict__ tgt,
                                float* __restrict__ out) {
    __shared__ float red[256];
    float s = 0.0f;
    for (int n = threadIdx.x; n < N_ROWS; n += 256) {
        float t  = tgt[n];
        float tc = fminf(fmaxf(t, -1.0f + EPS_CLIP), 1.0f - EPS_CLIP);
        // S * cos(acos(tc) + M) = S * (tc*cosM - sqrt(1-tc^2)*sinM)
        float num   = S_SCALE * (tc * COS_M - sqrtf(fmaxf(1.0f - tc * tc, 0.0f)) * SIN_M);
        float denom = __expf(num) + rowsum[n] - __expf(S_SCALE * t);
        s += num - __logf(denom);
    }
    red[threadIdx.x] = s;
    __syncthreads();
    #pragma unroll
    for (int off = 128; off > 0; off >>= 1) {
        if (threadIdx.x < off) red[threadIdx.x] += red[threadIdx.x + off];
        __syncthreads();
    }
    if (threadIdx.x == 0) out[0] = -red[0] / (float)N_ROWS;
}

extern "C" void kernel_launch(void* const* d_in, const int* in_sizes, int n_in,
                              void* d_out, int out_size, void* d_ws, size_t ws_size,
                              hipStream_t stream) {
    const float* x      = (const float*)d_in[0];
    const int*   labels = (const int*)d_in[1];
    const float* W      = (const float*)d_in[2];
    float* out = (float*)d_out;

    float* ws       = (float*)d_ws;
    float* inv_norm = ws;                 // [N]
    float* rowsum   = ws + N_ROWS;        // [N]
    float* tgt      = ws + 2 * N_ROWS;    // [N]

    init_rowsum_kernel<<<(N_ROWS + 255) / 256, 256, 0, stream>>>(rowsum);
    rownorm_kernel<<<(N_ROWS * 32) / 256, 256, 0, stream>>>(x, inv_norm);
    target_kernel<<<(N_ROWS * 32) / 256, 256, 0, stream>>>(x, W, labels, inv_norm, tgt);

    dim3 grid((C_DIM + BC - 1) / BC, N_ROWS / BN);
    gemm_exp_kernel<<<grid, 256, 0, stream>>>(x, W, inv_norm, rowsum);

    finalize_kernel<<<1, 256, 0, stream>>>(rowsum, tgt, out);
}